// PairwiseLoss_79104707657795
// MI455X (gfx1250) — compile-verified
//
#include <hip/hip_runtime.h>
#include <stdint.h>

// Problem shape (fixed by the reference): B=64, N=512, output = (B, N, N) f32.
#define B_DIM   64
#define N_DIM   512
#define TILE_J  8        // j-rows per block
#define THREADS 128      // 128 lanes * 4 k-values = 512 = N

typedef __attribute__((ext_vector_type(4))) float v4f;
typedef __attribute__((ext_vector_type(4))) int   v4i;

// 5 * log2(e): sigmoid(5d) = 1 / (1 + exp2(-5*log2e*d))
#define FIVE_LOG2E 7.2134752044448170367f

// Truncate a generic (flat) pointer to shared memory down to its 32-bit LDS
// byte offset (flat->LDS mapping discards the upper 32 aperture bits).
__device__ __forceinline__ uint32_t lds_off(const void* p) {
    return (uint32_t)(uintptr_t)p;
}

__global__ __launch_bounds__(THREADS)
void pairwise_loss_kernel(const float* __restrict__ out_in,
                          const int*   __restrict__ tgt_in,
                          const float* __restrict__ mask_in,
                          float*       __restrict__ loss_out)
{
    __shared__ __align__(16) float s_o[N_DIM];
    __shared__ __align__(16) float s_m[N_DIM];
    __shared__ __align__(16) int   s_t[N_DIM];

    const int tid    = threadIdx.x;
    const int jTiles = N_DIM / TILE_J;                 // 64
    const int b      = blockIdx.x / jTiles;
    const int jbase  = (blockIdx.x % jTiles) * TILE_J;

    // ---- Stage the three input rows for batch b into LDS with async
    // global->LDS B128 copies (CDNA5 path, tracked by ASYNCcnt). Each of the
    // 128 lanes moves 16 bytes per array: 128*16B = 2KB = one full row.
    {
        const uint32_t boff = (uint32_t)tid * 16u;
        const uint32_t lo = lds_off(&s_o[0]) + boff;
        const uint32_t lm = lds_off(&s_m[0]) + boff;
        const uint32_t lt = lds_off(&s_t[0]) + boff;
        const unsigned long long po =
            (unsigned long long)(uintptr_t)(out_in  + (size_t)b * N_DIM);
        const unsigned long long pm =
            (unsigned long long)(uintptr_t)(mask_in + (size_t)b * N_DIM);
        const unsigned long long pt =
            (unsigned long long)(uintptr_t)(tgt_in  + (size_t)b * N_DIM);
        // GVS mode: 64-bit SGPR base plus a per-lane 32-bit VGPR byte offset
        asm volatile("global_load_async_to_lds_b128 %0, %1, %2"
                     :: "v"(lo), "v"(boff), "s"(po) : "memory");
        asm volatile("global_load_async_to_lds_b128 %0, %1, %2"
                     :: "v"(lm), "v"(boff), "s"(pm) : "memory");
        asm volatile("global_load_async_to_lds_b128 %0, %1, %2"
                     :: "v"(lt), "v"(boff), "s"(pt) : "memory");
        asm volatile("s_wait_asynccnt 0" ::: "memory");
    }
    __syncthreads();

    // ---- Register-resident k-slice for this lane (one ds_load_b128 each).
    const int k0 = tid * 4;
    const v4f okv = *(const v4f*)&s_o[k0];
    const v4f mkv = *(const v4f*)&s_m[k0];
    const v4i tkv = *(const v4i*)&s_t[k0];

    const float okk[4] = {okv.x, okv.y, okv.z, okv.w};
    const float mkk[4] = {mkv.x, mkv.y, mkv.z, mkv.w};
    const int   tkk[4] = {tkv.x, tkv.y, tkv.z, tkv.w};

    float* outp = loss_out + ((size_t)b * N_DIM + (size_t)jbase) * N_DIM + k0;

    #pragma unroll
    for (int jj = 0; jj < TILE_J; ++jj) {
        const int   j  = jbase + jj;
        const float oj = s_o[j];   // uniform broadcast reads from LDS
        const float mj = s_m[j];
        const int   tj = s_t[j];

        v4f res;
        float r[4];
        #pragma unroll
        for (int i = 0; i < 4; ++i) {
            const float m = mj * mkk[i];
            const float d = oj - okk[i];
            // sigmoid(5d) via the hardware transcendental path:
            // one v_exp_f32 (exp2) + one v_rcp_f32, ~1 ulp each.
            const float e = __builtin_amdgcn_exp2f(-FIVE_LOG2E * d);
            float p = m * __builtin_amdgcn_rcpf(1.0f + e);
            const float traw = (tj > tkk[i]) ? 1.0f
                               : ((tj < tkk[i]) ? 0.0f : 0.5f);
            float t = traw * m;
            const float hm = (t != 0.5f) ? 1.0f : 0.0f;
            p *= hm;
            t *= hm;
            const float dd = p - t;
            r[i] = dd * dd * m;
        }
        res.x = r[0]; res.y = r[1]; res.z = r[2]; res.w = r[3];
        // Output is written once and never re-read by this kernel: NT store.
        __builtin_nontemporal_store(res, (v4f*)(outp + (size_t)jj * N_DIM));
    }
}

extern "C" void kernel_launch(void* const* d_in, const int* in_sizes, int n_in,
                              void* d_out, int out_size, void* d_ws, size_t ws_size,
                              hipStream_t stream)
{
    (void)in_sizes; (void)n_in; (void)out_size; (void)d_ws; (void)ws_size;
    const float* out_in  = (const float*)d_in[0];   // output  (B, N) f32
    const int*   tgt_in  = (const int*)  d_in[1];   // target  (B, N) int
    const float* mask_in = (const float*)d_in[2];   // mask    (B, N) f32
    float*       loss    = (float*)d_out;           // (B, N, N) f32

    const dim3 grid(B_DIM * (N_DIM / TILE_J));      // 4096 blocks
    const dim3 block(THREADS);                      // 128 threads = 4 waves
    pairwise_loss_kernel<<<grid, block, 0, stream>>>(out_in, tgt_in, mask_in, loss);
}